// CorrectCosMean_59734405152838
// MI455X (gfx1250) — compile-verified
//
#include <hip/hip_runtime.h>

typedef float f4  __attribute__((ext_vector_type(4)));
typedef float v2f __attribute__((ext_vector_type(2)));
typedef float v8f __attribute__((ext_vector_type(8)));

// ---------------------------------------------------------------------------
// Kernel 1: streaming passthrough copy of cos_t -> d_out[0 : N*C]
// Pure HBM-bandwidth bound: 128-bit per-lane loads/stores, non-temporal hints
// (TH=NT) so we don't thrash L2 with a 1.6 GB one-shot stream.
// ---------------------------------------------------------------------------
__global__ void cc_copy4_59734405152838(const f4* __restrict__ src, f4* __restrict__ dst,
                                        size_t n4,
                                        const float* __restrict__ srcs, float* __restrict__ dsts,
                                        size_t n) {
  const size_t stride = (size_t)gridDim.x * blockDim.x;
  for (size_t i = (size_t)blockIdx.x * blockDim.x + threadIdx.x; i < n4; i += stride) {
    f4 v = __builtin_nontemporal_load(src + i);
    __builtin_nontemporal_store(v, dst + i);
  }
  // scalar tail (n % 4), normally empty for N*C = 409,600,000
  if (blockIdx.x == 0 && threadIdx.x == 0) {
    for (size_t i = n4 * 4; i < n; ++i) dsts[i] = srcs[i];
  }
}

// ---------------------------------------------------------------------------
// Kernel 2: gather cos_t[i, y[i]] and deterministically reduce to 32 block
// partial sums in d_ws (no float atomics -> bitwise-reproducible).
// ---------------------------------------------------------------------------
__global__ void cc_gather_partial_59734405152838(const float* __restrict__ cos_t,
                                                 const long long* __restrict__ y,
                                                 float* __restrict__ ws,
                                                 int rows, int C) {
  const int tid = threadIdx.x;
  const int stride = gridDim.x * blockDim.x;
  float acc = 0.0f;
  for (int i = blockIdx.x * blockDim.x + tid; i < rows; i += stride) {
    const int label = (int)y[i];  // int64 label, value < 100000 fits in low word
    acc += cos_t[(size_t)i * (size_t)C + (size_t)label];
  }
  __shared__ float s[128];
  s[tid] = acc;
  __syncthreads();
  for (int off = 64; off > 0; off >>= 1) {
    if (tid < off) s[tid] += s[tid + off];
    __syncthreads();
  }
  if (tid == 0) ws[blockIdx.x] = s[0];
}

// ---------------------------------------------------------------------------
// Kernel 3: one wave32. Fold the 32 partials with V_WMMA_F32_16X16X4_F32.
// A layout (16x4 fp32): lane l, VGPR0 = A[l%16][l<16 ? 0 : 2], VGPR1 = K1/K3.
// We set a.x = ws[lane], a.y = 0  =>  A[M][0]=ws[M], A[M][2]=ws[M+16].
// B = ones(4x16)  =>  D[M][j] = ws[M] + ws[M+16] for every column j.
// D layout: lane 0 holds D[0..7][0], lane 16 holds D[8..15][0]; summing each
// lane's 8 accumulator VGPRs and adding lane0+lane16 gives the exact total.
// EXEC is all ones (32 threads, no divergence) as WMMA requires.
// ---------------------------------------------------------------------------
__global__ void cc_final_mean_59734405152838(const float* __restrict__ ws,
                                             float* __restrict__ out,
                                             float inv_rows) {
  const int lane = threadIdx.x;
  v2f a; a.x = ws[lane]; a.y = 0.0f;
  v2f b; b.x = 1.0f;     b.y = 1.0f;
  v8f c = {};
  v8f d = __builtin_amdgcn_wmma_f32_16x16x4_f32(
      /*neg_a=*/false, a, /*neg_b=*/false, b,
      /*c_mod=*/(short)0, c, /*reuse_a=*/false, /*reuse_b=*/false);
  float s = d[0] + d[1] + d[2] + d[3] + d[4] + d[5] + d[6] + d[7];
  __shared__ float sh[32];
  sh[lane] = s;
  __syncthreads();
  if (lane == 0) out[0] = (sh[0] + sh[16]) * inv_rows;
}

extern "C" void kernel_launch(void* const* d_in, const int* in_sizes, int n_in,
                              void* d_out, int out_size, void* d_ws, size_t ws_size,
                              hipStream_t stream) {
  const float*     cos_t = (const float*)d_in[0];
  const long long* y     = (const long long*)d_in[1];
  float* out = (float*)d_out;
  float* ws  = (float*)d_ws;

  const size_t n    = (size_t)in_sizes[0];          // N*C = 409,600,000
  const int    rows = in_sizes[1];                  // N = 4096
  const int    C    = (int)(n / (size_t)rows);      // 100,000

  // --- bulk copy: ~3.28 GB HBM traffic, the only thing that matters ---
  const size_t n4 = n / 4;
  unsigned copy_blocks = (unsigned)((n4 + 255) / 256);
  if (copy_blocks > 16384u) copy_blocks = 16384u;   // ~4.2M threads, ~24 iters each
  if (copy_blocks == 0u)    copy_blocks = 1u;
  cc_copy4_59734405152838<<<copy_blocks, 256, 0, stream>>>(
      (const f4*)cos_t, (f4*)out, n4, cos_t, out, n);

  // --- gather + deterministic reduction (negligible cost) ---
  cc_gather_partial_59734405152838<<<32, 128, 0, stream>>>(cos_t, y, ws, rows, C);
  cc_final_mean_59734405152838<<<1, 32, 0, stream>>>(ws, out + n, 1.0f / (float)rows);
}